// DotPredictor_23072564314868
// MI455X (gfx1250) — compile-verified
//
#include <hip/hip_runtime.h>

// CDNA5 / gfx1250: wave32, WMMA f32 16x16x4 (full fp32 precision — matches
// the fp32 reference arithmetic).

typedef float v2f __attribute__((ext_vector_type(2)));
typedef float v4f __attribute__((ext_vector_type(4)));
typedef float v8f __attribute__((ext_vector_type(8)));

#define D_FEAT 128

// One wave = 16 edges. D[i][j] = dot(hu_i, hv_j) accumulated over 32 WMMA
// chunks of K=4; keep the diagonal.
//
// Feature->chunk remap for wide loads: chunk kc covers features
//   K-slots 0,1 -> {2kc, 2kc+1}          (served by lanes 0..15,  half=0)
//   K-slots 2,3 -> {64+2kc, 64+2kc+1}    (served by lanes 16..31, half=1)
// Each lane's base is  h + node*128 + half*64, and one global_load_b128 at
// byte offset 16*j supplies chunks 2j (xy) and 2j+1 (zw). All 128 features
// are covered exactly once -> exact fp32 dot product.
//
// ISA VGPR layouts (CDNA5 ISA):
//   A (16x4 f32): lane m -> A[m][0..1], lane m+16 -> A[m][2..3]
//   B (4x16 f32): lane n -> B[0..1][n], lane n+16 -> B[2..3][n]
//   D (16x16 f32): diag D[m][m] in lanes 0..7 (vgpr=lane) and
//                  lanes 24..31 (vgpr=lane-24).

__global__ __launch_bounds__(256) void edge_dot_wmma_f32(
    const float* __restrict__ h,
    const int*   __restrict__ src,
    const int*   __restrict__ dst,
    float*       __restrict__ out,
    int n_edges)
{
  const int lane       = threadIdx.x & 31;
  const int wave_local = threadIdx.x >> 5;
  const int wave       = blockIdx.x * (blockDim.x >> 5) + wave_local;

  const int e0 = wave * 16;
  if (e0 >= n_edges) return;            // wave-uniform: EXEC stays all-1s

  const int m    = lane & 15;           // edge within the 16-edge tile
  const int half = lane >> 4;           // 0 -> features 0..63, 1 -> 64..127

  int e = e0 + m;
  if (e > n_edges - 1) e = n_edges - 1; // tail clamp (E%16==0 in practice)

  const size_t su = (size_t)src[e];
  const size_t du = (size_t)dst[e];

  const float* __restrict__ pu = h + su * D_FEAT + half * 64;
  const float* __restrict__ pv = h + du * D_FEAT + half * 64;

  v8f acc = {};
#pragma unroll 4
  for (int j = 0; j < 16; ++j) {
    v4f a = *(const v4f*)(pu + 4 * j);   // global_load_b128 (16B aligned)
    v4f b = *(const v4f*)(pv + 4 * j);

    v2f alo = __builtin_shufflevector(a, a, 0, 1);
    v2f ahi = __builtin_shufflevector(a, a, 2, 3);
    v2f blo = __builtin_shufflevector(b, b, 0, 1);
    v2f bhi = __builtin_shufflevector(b, b, 2, 3);

    // D = A x B + C, all fp32: (neg_a, A, neg_b, B, c_mod, C, reuse_a, reuse_b)
    acc = __builtin_amdgcn_wmma_f32_16x16x4_f32(
        false, alo, false, blo, (short)0, acc, false, false);
    acc = __builtin_amdgcn_wmma_f32_16x16x4_f32(
        false, ahi, false, bhi, (short)0, acc, false, false);
  }

  // Extract diagonal D[m][m].
  const int sel = (lane < 16) ? lane : (lane - 24);
  float r = acc[0];
#pragma unroll
  for (int i = 1; i < 8; ++i) r = (sel == i) ? acc[i] : r;

  if (lane < 8) {                       // D[m][m], m = 0..7
    const int eo = e0 + lane;
    if (eo < n_edges) out[eo] = r;
  } else if (lane >= 24) {              // D[m][m], m = 8..15
    const int eo = e0 + lane - 16;
    if (eo < n_edges) out[eo] = r;
  }
}

extern "C" void kernel_launch(void* const* d_in, const int* in_sizes, int n_in,
                              void* d_out, int out_size, void* d_ws, size_t ws_size,
                              hipStream_t stream) {
  (void)n_in; (void)d_ws; (void)ws_size; (void)out_size;

  const float* h   = (const float*)d_in[0];
  const int*   src = (const int*)d_in[1];
  const int*   dst = (const int*)d_in[2];
  float*       out = (float*)d_out;

  const int n_edges = in_sizes[1];              // 640000
  const int waves   = (n_edges + 15) / 16;      // 16 edges per wave
  const int blocks  = (waves + 7) / 8;          // 8 waves (256 thr) per block

  edge_dot_wmma_f32<<<blocks, 256, 0, stream>>>(h, src, dst, out, n_edges);
}